// Model_53970559042128
// MI455X (gfx1250) — compile-verified
//
#include <hip/hip_runtime.h>

// ---------------------------------------------------------------------------
// CDNA5 (gfx1250) GRU stack.  T=2048, B=128, H=16, G=48, fp32.
//   gru_proj_kernel<NKB> : xg = seq @ Wih^T + bih (parallel fp32 WMMA GEMM)
//   gru_scan_kernel      : sequential recurrence. Transposed WMMA
//                          (A = Whh tiles resident in VGPRs, B = h^T built
//                          per step with a half-wave __shfl_xor swap).
//                          No LDS round-trips, branch-free fast gate math.
// ---------------------------------------------------------------------------

typedef __attribute__((ext_vector_type(2))) float v2f;
typedef __attribute__((ext_vector_type(4))) float v4f;
typedef __attribute__((ext_vector_type(8))) float v8f;

__device__ __forceinline__ v8f wmma_f32_k4(v2f a, v2f b, v8f c) {
  // D(16x16,f32) = A(16x4,f32) * B(4x16,f32) + C
  return __builtin_amdgcn_wmma_f32_16x16x4_f32(
      false, a, false, b, (short)0, c, false, false);
}

// Branch-free fast nonlinearities on native v_exp_f32 / v_rcp_f32.
__device__ __forceinline__ float fast_sigmoid(float x) {
  return __builtin_amdgcn_rcpf(1.f + __expf(-x));
}
__device__ __forceinline__ float fast_tanh(float x) {
  // 1 - 2/(e^{2x}+1); saturates to +/-1 without branches.
  return 1.f - 2.f * __builtin_amdgcn_rcpf(__expf(2.f * x) + 1.f);
}

#define TT 2048
#define BB 128
#define KSTRIDE 33  // LDS row stride (max Kpad=32, +1 anti-bank-conflict)

// ---------------------------------------------------------------------------
// Input projection, transposed orientation per 16-row chunk:
//   D'_g[u, b] = sum_k wih[g*16+u, k] * in[row(b), k]   (+ bih)
// Grid: (T*B)/128 blocks x 256 threads (8 waves, wave = 16 batch rows).
// ---------------------------------------------------------------------------
template <int NKB>
__global__ void gru_proj_kernel(const float* __restrict__ in, long rs_t, long rs_b,
                                const float* __restrict__ wih,
                                const float* __restrict__ bih,
                                float* __restrict__ xg, int K) {
  const int Kpad = NKB * 4;
  __shared__ float Ilds[128 * KSTRIDE];
  __shared__ float Wlds[48 * KSTRIDE];
  const int tid = threadIdx.x;
  const long row0 = (long)blockIdx.x * 128;

  for (int i = tid; i < 128 * Kpad; i += 256) {
    int r = i / Kpad, k = i - r * Kpad;
    long gr = row0 + r;
    long t = gr >> 7;  // / B (B==128)
    long b = gr & 127;
    Ilds[r * KSTRIDE + k] = (k < K) ? in[t * rs_t + b * rs_b + k] : 0.f;
  }
  for (int i = tid; i < 48 * Kpad; i += 256) {
    int g = i / Kpad, k = i - g * Kpad;
    Wlds[g * KSTRIDE + k] = (k < K) ? wih[g * K + k] : 0.f;
  }
  __syncthreads();

  const int lane = tid & 31, wv = tid >> 5;
  const int hi = lane >> 4, ln = lane & 15;
  const int m0 = wv * 16;  // batch-row chunk of this wave

  // B fragments: B[k][col] = in_row(m0+col)[k]
  v2f bfr[NKB];
#pragma unroll
  for (int kb = 0; kb < NKB; ++kb) {
    int kk = kb * 4 + 2 * hi;
    bfr[kb].x = Ilds[(m0 + ln) * KSTRIDE + kk];
    bfr[kb].y = Ilds[(m0 + ln) * KSTRIDE + kk + 1];
  }

  float* op = xg + (row0 + m0 + ln) * 48 + 8 * hi;
#pragma unroll
  for (int g = 0; g < 3; ++g) {
    v8f c;
#pragma unroll
    for (int v = 0; v < 8; ++v) c[v] = bih ? bih[g * 16 + v + 8 * hi] : 0.f;
#pragma unroll
    for (int kb = 0; kb < NKB; ++kb) {
      v2f a;  // A[u][k] = wih[g*16+u][k], u = lane&15
      int kk = kb * 4 + 2 * hi;
      a.x = Wlds[(g * 16 + ln) * KSTRIDE + kk];
      a.y = Wlds[(g * 16 + ln) * KSTRIDE + kk + 1];
      c = wmma_f32_k4(a, bfr[kb], c);
    }
    v4f clo = {c[0], c[1], c[2], c[3]};
    v4f chi = {c[4], c[5], c[6], c[7]};
    *(v4f*)(op + g * 16) = clo;
    *(v4f*)(op + g * 16 + 4) = chi;
  }
}

// ---------------------------------------------------------------------------
// Sequential GRU scan. Grid = 4 blocks per direction (2 waves/block);
// each wave independently owns 16 batch columns for all T steps.
// ---------------------------------------------------------------------------
struct ScanArgs {
  const float* xg;   // (T*B) x 48, includes bih
  const float* whh;  // 48 x 16 row-major
  const float* bhh;  // 48 or null
  const float* h0;   // (B,16) or null (zeros)
  float* out;        // ys base
  float* hT;         // (B,16) final hidden
  long st_t, st_b;   // element strides of out
  int col_off;       // 0 fwd, 16 bwd (bidir concat)
  int t0, dt;        // time start / step (+1 / -1)
};
struct ScanLaunch { ScanArgs s[2]; };

__global__ void gru_scan_kernel(ScanLaunch L, int T) {
  const int bidx = blockIdx.x;
  const ScanArgs s = L.s[bidx >> 2];
  const int tid = threadIdx.x;
  const int lane = tid & 31, wv = tid >> 5;
  const int hi = lane >> 4, ln = lane & 15;
  const int b0 = ((bidx & 3) * 2 + wv) * 16;  // batch column base

  // A fragments: Whh gate tiles, time-invariant, resident in VGPRs.
  // A_g[u][k] = whh[g*16+u][k], u = lane&15.
  v2f afr[3][4];
#pragma unroll
  for (int g = 0; g < 3; ++g)
#pragma unroll
    for (int kb = 0; kb < 4; ++kb) {
      int kk = kb * 4 + 2 * hi;
      afr[g][kb].x = s.whh[(g * 16 + ln) * 16 + kk];
      afr[g][kb].y = s.whh[(g * 16 + ln) * 16 + kk + 1];
    }
  float bhv[3][8];
#pragma unroll
  for (int g = 0; g < 3; ++g)
#pragma unroll
    for (int v = 0; v < 8; ++v)
      bhv[g][v] = s.bhh ? s.bhh[g * 16 + v + 8 * hi] : 0.f;

  // h^T state in C/D layout: hct[v] = h[b0+ln][v + 8*hi]
  float hct[8];
  {
    const float* hp = s.h0 ? s.h0 + (long)(b0 + ln) * 16 + 8 * hi : nullptr;
#pragma unroll
    for (int v = 0; v < 8; ++v) hct[v] = hp ? hp[v] : 0.f;
  }

  const long xg_step = (long)s.dt * BB * 48;
  const float* __restrict__ xgp = s.xg + ((long)s.t0 * BB + b0 + ln) * 48;
  float* __restrict__ outp =
      s.out + (long)s.t0 * s.st_t + (long)(b0 + ln) * s.st_b + s.col_off + 8 * hi;
  const long out_step = (long)s.dt * s.st_t;

  // first tile (per lane: 6 x b128 contiguous loads)
  v4f xr0 = *(const v4f*)(xgp + 8 * hi);
  v4f xr1 = *(const v4f*)(xgp + 8 * hi + 4);
  v4f xz0 = *(const v4f*)(xgp + 16 + 8 * hi);
  v4f xz1 = *(const v4f*)(xgp + 16 + 8 * hi + 4);
  v4f xn0 = *(const v4f*)(xgp + 32 + 8 * hi);
  v4f xn1 = *(const v4f*)(xgp + 32 + 8 * hi + 4);

  for (int step = 0; step < T; ++step) {
    // double-buffer: issue next step's xg loads before compute
    const float* xq = xgp + ((step + 1 < T) ? xg_step : 0);
    v4f nr0 = *(const v4f*)(xq + 8 * hi);
    v4f nr1 = *(const v4f*)(xq + 8 * hi + 4);
    v4f nz0 = *(const v4f*)(xq + 16 + 8 * hi);
    v4f nz1 = *(const v4f*)(xq + 16 + 8 * hi + 4);
    v4f nn0 = *(const v4f*)(xq + 32 + 8 * hi);
    v4f nn1 = *(const v4f*)(xq + 32 + 8 * hi + 4);
    // pull future tiles HBM -> L2 (speculative)
    if (step + 8 < T) __builtin_prefetch((const char*)(xgp + 8 * xg_step), 0, 3);

    // B fragments of h^T via half-wave swap (no LDS round-trip):
    // B[k][col] at lane(ln,hi) needs rows k = kb*4+2*hi, +1.
    float sw[8];
#pragma unroll
    for (int v = 0; v < 8; ++v) sw[v] = __shfl_xor(hct[v], 16, 32);
    v2f bfr[4];
    bfr[0].x = hi ? sw[2] : hct[0];  bfr[0].y = hi ? sw[3] : hct[1];
    bfr[1].x = hi ? sw[6] : hct[4];  bfr[1].y = hi ? sw[7] : hct[5];
    bfr[2].x = hi ? hct[2] : sw[0];  bfr[2].y = hi ? hct[3] : sw[1];
    bfr[3].x = hi ? hct[6] : sw[4];  bfr[3].y = hi ? hct[7] : sw[5];

    // C init: r/z gates seeded with xg (includes bih); n gate with bhh_n.
    v8f cr, cz, cn;
    float xnv[8];
#pragma unroll
    for (int v = 0; v < 4; ++v) {
      cr[v] = xr0[v];     cr[v + 4] = xr1[v];
      cz[v] = xz0[v];     cz[v + 4] = xz1[v];
      cn[v] = bhv[2][v];  cn[v + 4] = bhv[2][v + 4];
      xnv[v] = xn0[v];    xnv[v + 4] = xn1[v];
    }

    // hg^T = Whh_g x h^T : 3 independent chains of 4 fp32 WMMAs
#pragma unroll
    for (int kb = 0; kb < 4; ++kb) {
      cr = wmma_f32_k4(afr[0][kb], bfr[kb], cr);
      cz = wmma_f32_k4(afr[1][kb], bfr[kb], cz);
      cn = wmma_f32_k4(afr[2][kb], bfr[kb], cn);
    }

    // gates + state update (branch-free, native exp/rcp)
    float hnew[8];
#pragma unroll
    for (int v = 0; v < 8; ++v) {
      float r = fast_sigmoid(cr[v] + bhv[0][v]);
      float z = fast_sigmoid(cz[v] + bhv[1][v]);
      float nn = fast_tanh(fmaf(r, cn[v], xnv[v]));
      hnew[v] = fmaf(z, hct[v] - nn, nn);
      hct[v] = hnew[v];
    }
    v4f olo = {hnew[0], hnew[1], hnew[2], hnew[3]};
    v4f ohi = {hnew[4], hnew[5], hnew[6], hnew[7]};
    *(v4f*)outp = olo;
    *(v4f*)(outp + 4) = ohi;

    xr0 = nr0; xr1 = nr1; xz0 = nz0; xz1 = nz1; xn0 = nn0; xn1 = nn1;
    xgp += xg_step;
    outp += out_step;
  }

  float* hp = s.hT + (long)(b0 + ln) * 16 + 8 * hi;
  v4f hlo = {hct[0], hct[1], hct[2], hct[3]};
  v4f hhi = {hct[4], hct[5], hct[6], hct[7]};
  *(v4f*)hp = hlo;
  *(v4f*)(hp + 4) = hhi;
}

// ---------------------------------------------------------------------------
// Host orchestration
// ---------------------------------------------------------------------------
struct P { const float* bhh; const float* bih; const float* whh; const float* wih; int isz; };

extern "C" void kernel_launch(void* const* d_in, const int* in_sizes, int n_in,
                              void* d_out, int out_size, void* d_ws, size_t ws_size,
                              hipStream_t stream) {
  (void)in_sizes; (void)n_in; (void)out_size; (void)ws_size;
  const int T = TT, B = BB;

  const float* x = (const float*)d_in[0];
  const float* y = (const float*)d_in[1];

  // JAX pytree flatten order: dicts sorted by key -> bhh, bih, whh, wih
  int idx = 2;
  auto getp_b = [&](int isz) {
    P p; p.bhh = (const float*)d_in[idx++]; p.bih = (const float*)d_in[idx++];
    p.whh = (const float*)d_in[idx++]; p.wih = (const float*)d_in[idx++];
    p.isz = isz; return p;
  };
  auto getp_n = [&](int isz) {
    P p; p.bhh = nullptr; p.bih = nullptr;
    p.whh = (const float*)d_in[idx++]; p.wih = (const float*)d_in[idx++];
    p.isz = isz; return p;
  };

  P p00 = getp_b(32);
  P p01[3]; for (int l = 0; l < 3; ++l) p01[l] = getp_n(16);
  P p02[8]; for (int l = 0; l < 4; ++l) for (int d = 0; d < 2; ++d) p02[l*2+d] = getp_b(l == 0 ? 16 : 32);
  P p03[8]; for (int l = 0; l < 4; ++l) for (int d = 0; d < 2; ++d) p03[l*2+d] = getp_b(l == 0 ? 16 : 32);
  P p10 = getp_b(25);
  P p11[3]; for (int l = 0; l < 3; ++l) p11[l] = getp_n(16);
  P p12[8]; for (int l = 0; l < 4; ++l) for (int d = 0; d < 2; ++d) p12[l*2+d] = getp_b(l == 0 ? 16 : 32);
  P p13[8]; for (int l = 0; l < 4; ++l) for (int d = 0; d < 2; ++d) p13[l*2+d] = getp_b(l == 0 ? 16 : 32);

  // d_out layout (floats): x2,x3,h0,h1,h2,h3,y2,y3,h4,h5,h6,h7
  float* OUT = (float*)d_out;
  const long SEQ32 = (long)T * B * 32;
  float* O_X2 = OUT;
  float* O_X3 = O_X2 + SEQ32;
  float* O_H0 = O_X3 + SEQ32;
  float* O_H1 = O_H0 + 1 * B * 16;
  float* O_H2 = O_H1 + 3 * B * 16;
  float* O_H3 = O_H2 + 8 * B * 16;
  float* O_Y2 = O_H3 + 8 * B * 16;
  float* O_Y3 = O_Y2 + SEQ32;
  float* O_H4 = O_Y3 + SEQ32;
  float* O_H5 = O_H4 + 1 * B * 16;
  float* O_H6 = O_H5 + 3 * B * 16;
  float* O_H7 = O_H6 + 8 * B * 16;

  // workspace
  float* ws = (float*)d_ws;
  size_t o = 0;
  float* XGf = ws + o; o += (size_t)T * B * 48;
  float* XGb = ws + o; o += (size_t)T * B * 48;
  float* SA  = ws + o; o += (size_t)T * B * 32;
  float* SB  = ws + o; o += (size_t)T * B * 32;
  float* X1  = ws + o; o += (size_t)T * B * 16;

  auto gemm = [&](const float* in, long rst, long rsb, const P& p, float* xgb) {
    dim3 g((T * B) / 128), b(256);
    if (p.isz == 16)
      hipLaunchKernelGGL(gru_proj_kernel<4>, g, b, 0, stream, in, rst, rsb, p.wih, p.bih, xgb, 16);
    else if (p.isz == 25)
      hipLaunchKernelGGL(gru_proj_kernel<7>, g, b, 0, stream, in, rst, rsb, p.wih, p.bih, xgb, 25);
    else
      hipLaunchKernelGGL(gru_proj_kernel<8>, g, b, 0, stream, in, rst, rsb, p.wih, p.bih, xgb, 32);
  };
  auto scan1 = [&](ScanArgs a) {
    ScanLaunch L; L.s[0] = a; L.s[1] = a;
    hipLaunchKernelGGL(gru_scan_kernel, dim3(4), dim3(64), 0, stream, L, T);
  };
  auto scan2 = [&](ScanArgs a, ScanArgs b) {
    ScanLaunch L; L.s[0] = a; L.s[1] = b;
    hipLaunchKernelGGL(gru_scan_kernel, dim3(8), dim3(64), 0, stream, L, T);
  };

  auto run_path = [&](const float* in0, long rs0_t, long rs0_b,
                      const P& q0, const P* q1, const P* q2, const P* q3,
                      float* O2, float* O3, long ost_t, long ost_b,
                      float* H0o, float* H1o, float* H2o, float* H3o) {
    // stack 0: 1 layer, bias
    gemm(in0, rs0_t, rs0_b, q0, XGf);
    scan1(ScanArgs{XGf, q0.whh, q0.bhh, nullptr, SA, H0o,
                   (long)(B * 16), 16L, 0, 0, +1});
    // stack 1: 3 layers, no bias: SA -> SB -> SA -> X1
    float* bufs[4] = {SA, SB, SA, X1};
    for (int l = 0; l < 3; ++l) {
      gemm(bufs[l], (long)(B * 16), 16L, q1[l], XGf);
      scan1(ScanArgs{XGf, q1[l].whh, q1[l].bhh, nullptr, bufs[l + 1],
                     H1o + (long)l * B * 16, (long)(B * 16), 16L, 0, 0, +1});
    }
    // stacks 2/3: 4 bidir layers each
    auto bidir = [&](const P* q, float* Hout, const float* Hinit, float* Ofin) {
      const float* pi = X1; long pt = B * 16, pb = 16;
      float* obufs[3] = {SA, SB, SA};
      for (int l = 0; l < 4; ++l) {
        gemm(pi, pt, pb, q[l * 2 + 0], XGf);
        gemm(pi, pt, pb, q[l * 2 + 1], XGb);
        float* ob; long st_t, st_b;
        if (l < 3) { ob = obufs[l]; st_t = B * 32; st_b = 32; }
        else       { ob = Ofin;     st_t = ost_t;  st_b = ost_b; }
        const float* h0f = Hinit ? Hinit + (long)(l * 2 + 0) * B * 16 : nullptr;
        const float* h0b = Hinit ? Hinit + (long)(l * 2 + 1) * B * 16 : nullptr;
        scan2(ScanArgs{XGf, q[l*2+0].whh, q[l*2+0].bhh, h0f, ob,
                       Hout + (long)(l*2+0) * B * 16, st_t, st_b, 0, 0, +1},
              ScanArgs{XGb, q[l*2+1].whh, q[l*2+1].bhh, h0b, ob,
                       Hout + (long)(l*2+1) * B * 16, st_t, st_b, 16, T - 1, -1});
        pi = ob; pt = st_t; pb = st_b;
      }
    };
    bidir(q2, H2o, nullptr, O2);
    bidir(q3, H3o, H2o, O3);  // p03/p13 seeded with h2/h6 (already in d_out)
  };

  // x path: x is (T,B,32); outputs (T,B,32)
  run_path(x, (long)B * 32, 32L, p00, p01, p02, p03,
           O_X2, O_X3, (long)B * 32, 32L, O_H0, O_H1, O_H2, O_H3);
  // y path: y is (B,T,25) -> read transposed; final outputs written as (B,T,32)
  run_path(y, 25L, (long)T * 25, p10, p11, p12, p13,
           O_Y2, O_Y3, 32L, (long)T * 32, O_H4, O_H5, O_H6, O_H7);
}